// GATLayer_53635551593170
// MI455X (gfx1250) — compile-verified
//
#include <hip/hip_runtime.h>
#include <hip/hip_bf16.h>
#include <stdint.h>

typedef __attribute__((ext_vector_type(2))) float v2f;
typedef __attribute__((ext_vector_type(8))) float v8f;
typedef __attribute__((ext_vector_type(2))) int   v2i;

#define N_NODES 8192
#define D_IN    512
#define D_OUT   64
#define LALPHA  0.2f

__device__ __forceinline__ v8f wmma_f32(v2f a, v2f b, v8f c) {
    // D = A(16x4 f32) * B(4x16 f32) + C(16x16 f32)
    return __builtin_amdgcn_wmma_f32_16x16x4_f32(false, a, false, b, (short)0, c,
                                                 false, false);
}

// CDNA5 async global->LDS copy (ASYNCcnt-tracked), 16B per lane.
// lds_off: byte offset in LDS aperture (addr[31:0] of a generic shared ptr).
__device__ __forceinline__ void async_load_b128(unsigned lds_off, const void* gptr) {
    asm volatile("global_load_async_to_lds_b128 %0, %1, off"
                 :
                 : "v"(lds_off), "v"((unsigned long long)(uintptr_t)gptr)
                 : "memory");
}
__device__ __forceinline__ unsigned lds_off_u32(const void* p) {
    return (unsigned)(uintptr_t)p;   // low 32 bits = LDS byte offset
}

// ---------------------------------------------------------------------------
// Kernel 1: pack W (512x64) into WMMA B-operand layout.
// Wpack[(c*4+nt)*64 + 2*l + v] = W[(4c + koff(l) + v)*64 + 16nt + (l%16)]
// ---------------------------------------------------------------------------
__global__ void pack_w_kernel(const float* __restrict__ W, float* __restrict__ Wpack) {
    int t = blockIdx.x * 256 + threadIdx.x;           // 32768 total
    if (t >= 32768) return;
    int slot = t >> 6;          // c*4 + nt
    int r    = t & 63;
    int l = r >> 1, v = r & 1;
    int c  = slot >> 2;
    int nt = slot & 3;
    int koff = (l >= 16) ? 2 : 0;
    int k = 4 * c + koff + v;
    int n = 16 * nt + (l & 15);
    Wpack[t] = W[k * D_OUT + n];
}

// ---------------------------------------------------------------------------
// Kernel 2: Wh = h @ W  (8192x512 @ 512x64) via V_WMMA_F32_16X16X4_F32.
// Block = 128 thr (4 waves), 16 rows per block; wave nt owns cols 16nt..
// h row-tile staged in LDS with stride 516 (conflict-free b64 A reads).
// Writes Wh row-major and Bpack (B-layout of Wh for the attention GEMM).
// ---------------------------------------------------------------------------
__global__ __launch_bounds__(128) void wh_kernel(const float* __restrict__ h,
                                                 const float* __restrict__ Wpack,
                                                 float* __restrict__ Wh,
                                                 float* __restrict__ Bpack) {
    __shared__ __align__(16) float hs[16 * 516];
    int t = threadIdx.x;
    int b = blockIdx.x;                               // row tile (16 rows)

    const float4* hg = (const float4*)(h + (size_t)b * 16 * D_IN);
    for (int q = t; q < 16 * 128; q += 128) {         // 2048 float4
        int row = q >> 7;
        int c4  = q & 127;
        float4 val = hg[row * 128 + c4];
        float* dst = &hs[row * 516 + 4 * c4];
        dst[0] = val.x; dst[1] = val.y; dst[2] = val.z; dst[3] = val.w;
    }
    __syncthreads();

    int l    = t & 31;
    int nt   = t >> 5;
    int m16  = l & 15;
    int half = l >> 4;
    int koff = 2 * half;

    v8f acc = {};
    for (int c = 0; c < 128; ++c) {
        v2f av = *(const v2f*)&hs[m16 * 516 + 4 * c + koff];
        v2f bv = *(const v2f*)&Wpack[(c * 4 + nt) * 64 + 2 * l];
        acc = wmma_f32(av, bv, acc);
    }

#pragma unroll
    for (int r = 0; r < 8; ++r) {
        int m = r + 8 * half;
        float val = acc[r];
        Wh[((size_t)b * 16 + m) * D_OUT + nt * 16 + m16] = val;
        // Bpack[((J*4+nt)*4 + c)*64 + 2*l' + v], J = b
        int c  = m >> 2;
        int hi = (m & 2) >> 1;
        int v  = m & 1;
        int lp = m16 + 16 * hi;
        Bpack[(((size_t)b * 4 + nt) * 4 + c) * 64 + 2 * lp + v] = val;
    }
}

// ---------------------------------------------------------------------------
// Kernel 3: s1 = Wh@a1, s2 = Wh@a2. One wave per row, shuffle reduction.
// ---------------------------------------------------------------------------
__global__ __launch_bounds__(256) void s_kernel(const float* __restrict__ Wh,
                                                const float* __restrict__ a,
                                                float* __restrict__ s1,
                                                float* __restrict__ s2) {
    int wv = threadIdx.x >> 5, l = threadIdx.x & 31;
    int row = blockIdx.x * 8 + wv;
    float w0 = Wh[(size_t)row * 64 + l];
    float w1 = Wh[(size_t)row * 64 + 32 + l];
    float p1 = w0 * a[l]      + w1 * a[32 + l];
    float p2 = w0 * a[64 + l] + w1 * a[96 + l];
    for (int off = 16; off; off >>= 1) {
        p1 += __shfl_xor(p1, off);
        p2 += __shfl_xor(p2, off);
    }
    if (l == 0) { s1[row] = p1; s2[row] = p2; }
}

// ---------------------------------------------------------------------------
// Kernel 4: global max of s2 (fixed softmax shift bound).
// ---------------------------------------------------------------------------
__global__ void s2max_kernel(const float* __restrict__ s2, float* __restrict__ out) {
    __shared__ float red[256];
    int t = threadIdx.x;
    float m = -3.0e38f;
    for (int q = t; q < N_NODES; q += 256) m = fmaxf(m, s2[q]);
    red[t] = m;
    __syncthreads();
    for (int s = 128; s; s >>= 1) {
        if (t < s) red[t] = fmaxf(red[t], red[t + s]);
        __syncthreads();
    }
    if (t == 0) out[0] = red[0];
}

// ---------------------------------------------------------------------------
// Kernel 5: out = elu(softmax(mask(leakyrelu(s1⊕s2), adj)) @ Wh + bias)
// Block = 256 thr (8 waves), 16 output rows per block. Wave w owns j-tiles
// J ≡ w (mod 8): fixed-shift softmax => partial sums are additive.
// adj tile: GLOBAL_LOAD_ASYNC_TO_LDS_B128 x2 (double buffered, ASYNCcnt) into
// stride-20-padded LDS -> conflict-free b64 A-layout reads -> WMMA.
// ---------------------------------------------------------------------------
__global__ __launch_bounds__(256) void attn_kernel(const int* __restrict__ adj,
                                                   const float* __restrict__ Bpack,
                                                   const float* __restrict__ s1g,
                                                   const float* __restrict__ s2g,
                                                   const float* __restrict__ s2maxp,
                                                   const float* __restrict__ bias,
                                                   float* __restrict__ out) {
    __shared__ __align__(16) int   adjbuf[8][2][16 * 20];   // 20 KB
    __shared__ float accw[8][16 * 64];                      // 32 KB
    __shared__ float lrow[16];

    int t = threadIdx.x, l = t & 31, w = t >> 5;
    int ib   = blockIdx.x;
    int m16  = l & 15;
    int half = l >> 4;
    int koff = 2 * half;

    float s1v   = s1g[ib * 16 + m16];
    float s2max = s2maxp[0];
    float eM = s1v + s2max;
    float Mv = eM > 0.f ? eM : LALPHA * eM;                 // per-row safe bound

    v8f acc0 = {}, acc1 = {}, acc2 = {}, acc3 = {};
    float lsum = 0.f;

    const int* adjr = adj + (size_t)ib * 16 * N_NODES;
    int lr  = l >> 2;                                       // 0..7 (tile row)
    int lc4 = (l & 3) * 4;                                  // col within tile

    unsigned boff0 = lds_off_u32(&adjbuf[w][0][0]);
    unsigned boff1 = lds_off_u32(&adjbuf[w][1][0]);
    unsigned ld0 = (unsigned)(lr * 20 + lc4) * 4u;          // lane LDS byte offs
    unsigned ld1 = (unsigned)((lr + 8) * 20 + lc4) * 4u;

    // prefetch first tile (J = w) into buffer 0
    async_load_b128(boff0 + ld0, adjr + (size_t)lr       * N_NODES + w * 16 + lc4);
    async_load_b128(boff0 + ld1, adjr + (size_t)(lr + 8) * N_NODES + w * 16 + lc4);

    int J = w;
    for (int it = 0; it < 64; ++it) {                       // 512 tiles / 8 waves
        int pp = it & 1;
        // prefetch next tile (wrapped index => always in-bounds, branch-free)
        int Jc = (J + 8) & 511;
        unsigned nb = pp ? boff0 : boff1;
        async_load_b128(nb + ld0, adjr + (size_t)lr       * N_NODES + Jc * 16 + lc4);
        async_load_b128(nb + ld1, adjr + (size_t)(lr + 8) * N_NODES + Jc * 16 + lc4);
        // current tile resident once only the 2 just-issued remain outstanding
        asm volatile("s_wait_asynccnt 0x2" ::: "memory");

        const int* buf = &adjbuf[w][pp][0];
        v2f p[4];
#pragma unroll
        for (int c = 0; c < 4; ++c) {                       // A-layout from LDS
            v2i av  = *(const v2i*)&buf[m16 * 20 + 4 * c + koff];
            v2f s2p = *(const v2f*)(s2g + J * 16 + 4 * c + koff);
            float e0 = s1v + s2p.x; e0 = e0 > 0.f ? e0 : LALPHA * e0;
            float e1 = s1v + s2p.y; e1 = e1 > 0.f ? e1 : LALPHA * e1;
            float p0 = (av.x > 0) ? __expf(e0 - Mv) : 0.f;
            float p1 = (av.y > 0) ? __expf(e1 - Mv) : 0.f;
            p[c].x = p0; p[c].y = p1;
            lsum += p0 + p1;
        }

        const float* bp = Bpack + (size_t)J * 16 * 64;      // J's 16 packed rows
#pragma unroll
        for (int c = 0; c < 4; ++c) {
            v2f b0 = *(const v2f*)&bp[(0 * 4 + c) * 64 + 2 * l];
            v2f b1 = *(const v2f*)&bp[(1 * 4 + c) * 64 + 2 * l];
            v2f b2 = *(const v2f*)&bp[(2 * 4 + c) * 64 + 2 * l];
            v2f b3 = *(const v2f*)&bp[(3 * 4 + c) * 64 + 2 * l];
            acc0 = wmma_f32(p[c], b0, acc0);
            acc1 = wmma_f32(p[c], b1, acc1);
            acc2 = wmma_f32(p[c], b2, acc2);
            acc3 = wmma_f32(p[c], b3, acc3);
        }
        J += 8;
    }
    asm volatile("s_wait_asynccnt 0x0" ::: "memory");       // drain trailing prefetch

    // per-wave partials -> LDS
    float* aw = &accw[w][0];
#pragma unroll
    for (int r = 0; r < 8; ++r) {
        int m = r + 8 * half;
        aw[m * 64 +  0 + m16] = acc0[r];
        aw[m * 64 + 16 + m16] = acc1[r];
        aw[m * 64 + 32 + m16] = acc2[r];
        aw[m * 64 + 48 + m16] = acc3[r];
    }
    if (t < 16) lrow[t] = 0.f;
    __syncthreads();
    atomicAdd(&lrow[m16], lsum);                            // lane covers row m16
    __syncthreads();

    for (int q = t; q < 1024; q += 256) {
        int row = q >> 6, col = q & 63;
        float s = 0.f;
#pragma unroll
        for (int ww = 0; ww < 8; ++ww) s += accw[ww][q];
        float li  = fmaxf(lrow[row], 1e-30f);
        float val = s / li + bias[col];
        out[((size_t)ib * 16 + row) * 64 + col] = val > 0.f ? val : __expf(val) - 1.f;
    }
}

// ---------------------------------------------------------------------------
extern "C" void kernel_launch(void* const* d_in, const int* in_sizes, int n_in,
                              void* d_out, int out_size, void* d_ws, size_t ws_size,
                              hipStream_t stream) {
    (void)in_sizes; (void)n_in; (void)out_size; (void)ws_size;
    const float* h    = (const float*)d_in[0];
    const int*   adj  = (const int*)d_in[1];
    const float* W    = (const float*)d_in[2];
    const float* a    = (const float*)d_in[3];
    const float* bias = (const float*)d_in[4];
    float* out = (float*)d_out;

    float* ws    = (float*)d_ws;
    float* Wpack = ws;                      //  32768 f (128 KB)
    float* Wh    = Wpack + 32768;           // 524288 f (2 MB)
    float* Bpack = Wh + 524288;             // 524288 f (2 MB)
    float* s1    = Bpack + 524288;          //   8192 f
    float* s2    = s1 + 8192;               //   8192 f
    float* s2mx  = s2 + 8192;               //      1 f

    pack_w_kernel<<<128, 256, 0, stream>>>(W, Wpack);
    wh_kernel<<<512, 128, 0, stream>>>(h, Wpack, Wh, Bpack);
    s_kernel<<<1024, 256, 0, stream>>>(Wh, a, s1, s2);
    s2max_kernel<<<1, 256, 0, stream>>>(s2, s2mx);
    attn_kernel<<<512, 256, 0, stream>>>(adj, Bpack, s1, s2, s2mx, bias, out);
}